// TokenPoolerTopK_57329223467565
// MI455X (gfx1250) — compile-verified
//
#include <hip/hip_runtime.h>

#define DIMN 512
#define POOLN 64
#define TOPKN 50
#define BN 4
#define TN 1024

typedef float v2f __attribute__((ext_vector_type(2)));
typedef float v4f __attribute__((ext_vector_type(4)));
typedef float v8f __attribute__((ext_vector_type(8)));

// ---------------------------------------------------------------------------
// Stage 1: y[r, 0:64] = LayerNorm(ReLU(x[r, 0:512] @ w1 + b1)) for r in [0,4096)
// One wave (32 threads) computes a 16-row x 64-col tile with f32 WMMA.
// ---------------------------------------------------------------------------
__global__ __launch_bounds__(32)
void tp_stage1(const float* __restrict__ x, const float* __restrict__ w1,
               const float* __restrict__ b1, const float* __restrict__ g1,
               const float* __restrict__ beta1, float* __restrict__ y)
{
    __shared__ float ylds[16][POOLN + 4];      // padded rows -> no bank conflicts
    const int lane = threadIdx.x;              // 0..31 (wave32)
    const int l = lane & 15;
    const int h = lane >> 4;
    const int rowBase = blockIdx.x * 16;

    v8f c0 = {}, c1 = {}, c2 = {}, c3 = {};
    const float* xrow = x + (size_t)(rowBase + l) * DIMN;

    #pragma unroll 4
    for (int kk = 0; kk < DIMN / 4; ++kk) {
        const int k0 = 4 * kk + 2 * h;
        // A 16x4 fp32: lane l holds (M=l, K=k0), (M=l, K=k0+1)
        v2f a = *(const v2f*)(xrow + k0);
        const float* wk = w1 + (size_t)k0 * POOLN + l;   // w1[k][p], row-major
        v2f b;
        b.x = wk[0];       b.y = wk[POOLN];
        c0 = __builtin_amdgcn_wmma_f32_16x16x4_f32(false, a, false, b, (short)0, c0, false, false);
        b.x = wk[16];      b.y = wk[POOLN + 16];
        c1 = __builtin_amdgcn_wmma_f32_16x16x4_f32(false, a, false, b, (short)0, c1, false, false);
        b.x = wk[32];      b.y = wk[POOLN + 32];
        c2 = __builtin_amdgcn_wmma_f32_16x16x4_f32(false, a, false, b, (short)0, c2, false, false);
        b.x = wk[48];      b.y = wk[POOLN + 48];
        c3 = __builtin_amdgcn_wmma_f32_16x16x4_f32(false, a, false, b, (short)0, c3, false, false);
    }

    // bias + ReLU into LDS (C/D layout: VGPR i -> row i+8h, col l within tile)
    #pragma unroll
    for (int i = 0; i < 8; ++i) {
        const int m = i + 8 * h;
        ylds[m][l]      = fmaxf(c0[i] + b1[l],      0.f);
        ylds[m][l + 16] = fmaxf(c1[i] + b1[l + 16], 0.f);
        ylds[m][l + 32] = fmaxf(c2[i] + b1[l + 32], 0.f);
        ylds[m][l + 48] = fmaxf(c3[i] + b1[l + 48], 0.f);
    }
    __syncthreads();

    // LayerNorm over POOL=64 per row: lane (l,h) handles row l, cols h*32..h*32+31
    float s = 0.f, s2 = 0.f;
    #pragma unroll
    for (int j = 0; j < 32; ++j) {
        float v = ylds[l][h * 32 + j];
        s += v; s2 += v * v;
    }
    s  += __shfl_xor(s,  16, 32);
    s2 += __shfl_xor(s2, 16, 32);
    const float mean = s * (1.f / POOLN);
    const float var  = s2 * (1.f / POOLN) - mean * mean;
    const float rstd = rsqrtf(var + 1e-5f);
    float* yout = y + (size_t)(rowBase + l) * POOLN;
    #pragma unroll
    for (int j = 0; j < 32; ++j) {
        const int p = h * 32 + j;
        yout[p] = (ylds[l][p] - mean) * rstd * g1[p] + beta1[p];
    }
}

// ---------------------------------------------------------------------------
// Stage 2: summed[t,:] = sum_b y[b,t,:]; idx[t,0:50] = top_k indices (desc,
// ties -> lowest index, matching jax.lax.top_k). One thread per t.
// ---------------------------------------------------------------------------
__global__ __launch_bounds__(256)
void tp_topk(const float* __restrict__ y, int* __restrict__ idx)
{
    const int t = blockIdx.x * blockDim.x + threadIdx.x;
    if (t >= TN) return;
    float s[POOLN];
    #pragma unroll
    for (int p = 0; p < POOLN; ++p) {
        s[p] = y[((size_t)0 * TN + t) * POOLN + p]
             + y[((size_t)1 * TN + t) * POOLN + p]
             + y[((size_t)2 * TN + t) * POOLN + p]
             + y[((size_t)3 * TN + t) * POOLN + p];
    }
    unsigned long long used = 0ull;
    #pragma unroll 1
    for (int k = 0; k < TOPKN; ++k) {
        float best = -3.402823e38f;
        int bi = 0;
        #pragma unroll
        for (int p = 0; p < POOLN; ++p) {
            const bool free = ((used >> p) & 1ull) == 0ull;
            if (free && s[p] > best) { best = s[p]; bi = p; }
        }
        used |= (1ull << bi);
        idx[t * TOPKN + k] = bi;
    }
}

// ---------------------------------------------------------------------------
// Stage 3: out[b,t,k,:] = LayerNorm(ReLU(y[b, idx[t,k], :] @ w2 + b2))
// 256 threads/block; 16 gathered rows per block; 8 waves x 4 N-tiles = N=512.
// Output-store bound -> non-temporal stores to keep w2/y resident in L2.
// ---------------------------------------------------------------------------
__global__ __launch_bounds__(256)
void tp_stage3(const float* __restrict__ y, const int* __restrict__ idx,
               const float* __restrict__ w2, const float* __restrict__ b2,
               const float* __restrict__ g2, const float* __restrict__ beta2,
               float* __restrict__ out)
{
    __shared__ float alds[16][POOLN + 4];   // 16 x 68 (pad: conflict-free A reads)
    __shared__ float olds[16][DIMN + 4];    // 16 x 516 (pad: conflict-free)
    const int tid = threadIdx.x;
    const int rowBase = blockIdx.x * 16;

    // Gather 16 A-rows: row r=(b*T+t)*TOPK+k  ->  y[b, idx[t,k], :]
    {
        const int m = tid >> 4;            // 0..15
        const int q = tid & 15;            // float4 chunk within the 64-row
        const int r = rowBase + m;
        const int b   = r / (TN * TOPKN);
        const int rem = r % (TN * TOPKN);
        const int t   = rem / TOPKN;
        const int k   = rem % TOPKN;
        const int src = idx[t * TOPKN + k];
        const float* yr = y + ((size_t)b * TN + src) * POOLN;
        *(v4f*)&alds[m][q * 4] = *(const v4f*)(yr + q * 4);
    }
    __syncthreads();

    const int lane = tid & 31;
    const int wv   = tid >> 5;             // 8 waves
    const int l = lane & 15;
    const int h = lane >> 4;
    const int colBase = wv * 64;           // this wave's 64 output columns

    v8f c0 = {}, c1 = {}, c2 = {}, c3 = {};
    #pragma unroll
    for (int kk = 0; kk < POOLN / 4; ++kk) {
        const int k0 = 4 * kk + 2 * h;
        v2f a = *(const v2f*)&alds[l][k0];
        const float* wk = w2 + (size_t)k0 * DIMN + colBase + l;  // w2[k][n]
        v2f b;
        b.x = wk[0];       b.y = wk[DIMN];
        c0 = __builtin_amdgcn_wmma_f32_16x16x4_f32(false, a, false, b, (short)0, c0, false, false);
        b.x = wk[16];      b.y = wk[DIMN + 16];
        c1 = __builtin_amdgcn_wmma_f32_16x16x4_f32(false, a, false, b, (short)0, c1, false, false);
        b.x = wk[32];      b.y = wk[DIMN + 32];
        c2 = __builtin_amdgcn_wmma_f32_16x16x4_f32(false, a, false, b, (short)0, c2, false, false);
        b.x = wk[48];      b.y = wk[DIMN + 48];
        c3 = __builtin_amdgcn_wmma_f32_16x16x4_f32(false, a, false, b, (short)0, c3, false, false);
    }

    // bias + ReLU into LDS out-tile
    #pragma unroll
    for (int i = 0; i < 8; ++i) {
        const int m = i + 8 * h;
        const int c = colBase + l;
        olds[m][c]      = fmaxf(c0[i] + b2[c],      0.f);
        olds[m][c + 16] = fmaxf(c1[i] + b2[c + 16], 0.f);
        olds[m][c + 32] = fmaxf(c2[i] + b2[c + 32], 0.f);
        olds[m][c + 48] = fmaxf(c3[i] + b2[c + 48], 0.f);
    }
    __syncthreads();

    // LayerNorm over DIM=512 per row: 16 threads per row, 32 cols each.
    const int row = tid >> 4;
    const int seg = tid & 15;
    float s = 0.f, s2 = 0.f;
    #pragma unroll
    for (int j = 0; j < 32; ++j) {
        const float v = olds[row][seg * 32 + j];
        s += v; s2 += v * v;
    }
    // 16-thread group is a contiguous half-wave -> xor masks 1,2,4,8 stay inside
    s  += __shfl_xor(s,  1, 32);  s2 += __shfl_xor(s2, 1, 32);
    s  += __shfl_xor(s,  2, 32);  s2 += __shfl_xor(s2, 2, 32);
    s  += __shfl_xor(s,  4, 32);  s2 += __shfl_xor(s2, 4, 32);
    s  += __shfl_xor(s,  8, 32);  s2 += __shfl_xor(s2, 8, 32);
    const float mean = s * (1.f / DIMN);
    const float var  = s2 * (1.f / DIMN) - mean * mean;
    const float rstd = rsqrtf(var + 1e-5f);

    float* orow = out + (size_t)(rowBase + row) * DIMN + seg * 32;
    #pragma unroll
    for (int j4 = 0; j4 < 8; ++j4) {
        v4f v  = *(const v4f*)&olds[row][seg * 32 + j4 * 4];
        v4f gg = *(const v4f*)(g2    + seg * 32 + j4 * 4);
        v4f bb = *(const v4f*)(beta2 + seg * 32 + j4 * 4);
        v4f o  = (v - mean) * rstd * gg + bb;
        __builtin_nontemporal_store(o, (v4f*)orow + j4);   // 400MB stream: bypass cache pressure
    }
}

// ---------------------------------------------------------------------------
extern "C" void kernel_launch(void* const* d_in, const int* in_sizes, int n_in,
                              void* d_out, int out_size, void* d_ws, size_t ws_size,
                              hipStream_t stream) {
    const float* x     = (const float*)d_in[0];
    const float* w1    = (const float*)d_in[1];
    const float* b1    = (const float*)d_in[2];
    const float* g1    = (const float*)d_in[3];
    const float* beta1 = (const float*)d_in[4];
    const float* w2    = (const float*)d_in[5];
    const float* b2    = (const float*)d_in[6];
    const float* g2    = (const float*)d_in[7];
    const float* beta2 = (const float*)d_in[8];
    float* out = (float*)d_out;

    // workspace: y [B*T, POOL] fp32, then idx [T, TOP_K] i32
    float* y  = (float*)d_ws;
    int*  idx = (int*)((char*)d_ws + (size_t)BN * TN * POOLN * sizeof(float));

    tp_stage1<<<(BN * TN) / 16, 32, 0, stream>>>(x, w1, b1, g1, beta1, y);
    tp_topk  <<<TN / 256, 256, 0, stream>>>(y, idx);
    tp_stage3<<<(BN * TN * TOPKN) / 16, 256, 0, stream>>>(y, idx, w2, b2, g2, beta2, out);
}